// Attention_84645215470257
// MI455X (gfx1250) — compile-verified
//
#include <hip/hip_runtime.h>
#include <math.h>

typedef float v2f __attribute__((ext_vector_type(2)));
typedef float v8f __attribute__((ext_vector_type(8)));

#define KNN     16
#define DM      64
#define DA      32
#define NHEAD   4
#define NCH     96           // 32 K-channels + 64 V-channels (logical)
#define NCHP    100          // LDS row stride (pad: 100 mod 64 = 36 -> conflict-free column walk)
#define NTILE   6            // 96 / 16 output tiles
#define NCHUNK  17           // ceil(67/4), K padded to 68
#define WPB     4            // waves per block
#define BLOCK   (WPB * 32)

__global__ __launch_bounds__(BLOCK) void knn_attn_kernel(
    const float* __restrict__ coords0, const float* __restrict__ coords1,
    const float* __restrict__ feats0,  const float* __restrict__ feats1,
    const int*   __restrict__ knn,
    const float* __restrict__ Wq, const float* __restrict__ bq,
    const float* __restrict__ Wk, const float* __restrict__ bk,
    const float* __restrict__ Wv, const float* __restrict__ bv,
    float* __restrict__ out, int n1)
{
    // ---- LDS ----
    __shared__ float WT[NCHUNK * NTILE * 32 * 2]; // pre-swizzled B fragments (float2/lane)
    __shared__ float WQs[DM * DA];                // Wq staged
    __shared__ float KV[WPB][KNN * NCHP];         // per-wave K|V result tile
    __shared__ float QB[WPB][DA];                 // per-wave q vector
    __shared__ float PB[WPB][KNN * NHEAD];        // per-wave softmax probs

    const int tid  = threadIdx.x;
    const int lane = tid & 31;
    const int w    = tid >> 5;

    // ---- stage weights into WMMA B-fragment layout ----
    // Assumed B layout for V_WMMA_F32_16X16X4_F32 (4x16, K x N), mirroring the
    // documented A 16x4 layout: lanes 0-15 hold rows K=0,1 in v0,v1; lanes
    // 16-31 hold rows K=2,3.  WT linear index = ((c*NTILE+t)*32 + lane)*2 + s.
    for (int i = tid; i < NCHUNK * NTILE * 32 * 2; i += BLOCK) {
        int s  = i & 1;
        int r  = i >> 1;
        int ln = r & 31;
        int r2 = r >> 5;
        int t  = r2 % NTILE;
        int c  = r2 / NTILE;
        int j  = 4 * c + 2 * (ln >> 4) + s;   // input row (0..67, 67 = zero pad)
        int ch = 16 * t + (ln & 15);          // fused output channel (0..95)
        float v = 0.f;
        if (j < 67) v = (ch < 32) ? Wk[j * DA + ch] : Wv[j * DM + (ch - 32)];
        WT[i] = v;
    }
    for (int i = tid; i < DM * DA; i += BLOCK) WQs[i] = Wq[i];
    __syncthreads();

    const int mrow = lane & 15;   // neighbor row / N-column within a tile
    const int ksel = lane >> 4;   // which half of the K-chunk this lane carries

    // bias per (lane, tile): independent of the point, hoist out of the loop
    float biasv[NTILE];
#pragma unroll
    for (int t = 0; t < NTILE; ++t) {
        int ch = 16 * t + mrow;
        biasv[t] = (ch < 32) ? bk[ch] : bv[ch - 32];
    }

    const int nwaves = (gridDim.x * blockDim.x) >> 5;
    const int wgid   = (blockIdx.x * blockDim.x + tid) >> 5;

    for (int n = wgid; n < n1; n += nwaves) {
        // ---- gather setup: this lane owns neighbor row `mrow` ----
        const int idxr = knn[n * KNN + mrow];
        const float* frow = feats0 + (size_t)idxr * DM;
        const float rx = coords0[(size_t)idxr * 3 + 0] - coords1[n * 3 + 0];
        const float ry = coords0[(size_t)idxr * 3 + 1] - coords1[n * 3 + 1];
        const float rz = coords0[(size_t)idxr * 3 + 2] - coords1[n * 3 + 2];

        // ---- Q projection (tiny: 64x32 per point) in fp32 VALU, lane -> q[lane]
        float qv = bq[lane];
        const float* f1row = feats1 + (size_t)n * DM;
        for (int j = 0; j < DM; ++j) qv += f1row[j] * WQs[j * DA + lane];
        QB[w][lane] = qv;

        // ---- fused K|V projection: (16x68) @ (68x96) via V_WMMA_F32_16X16X4_F32
        v8f acc[NTILE];
        const v8f zero8 = {0.f, 0.f, 0.f, 0.f, 0.f, 0.f, 0.f, 0.f};
#pragma unroll
        for (int t = 0; t < NTILE; ++t) acc[t] = zero8;

        const float2* frow2 = (const float2*)frow;
        for (int c = 0; c < 16; ++c) {                  // K-chunks over feats (cols 0..63)
            float2 af = frow2[2 * c + ksel];            // contiguous pair gather
            v2f a; a.x = af.x; a.y = af.y;
#pragma unroll
            for (int t = 0; t < NTILE; ++t) {
                float2 bf = ((const float2*)WT)[(c * NTILE + t) * 32 + lane];
                v2f b; b.x = bf.x; b.y = bf.y;
                acc[t] = __builtin_amdgcn_wmma_f32_16x16x4_f32(
                    false, a, false, b, (short)0, acc[t], false, false);
            }
        }
        {   // final chunk: cols 64..67 = [rel.x rel.y rel.z 0]
            v2f a;
            a.x = ksel ? rz : rx;
            a.y = ksel ? 0.f : ry;
#pragma unroll
            for (int t = 0; t < NTILE; ++t) {
                float2 bf = ((const float2*)WT)[(16 * NTILE + t) * 32 + lane];
                v2f b; b.x = bf.x; b.y = bf.y;
                acc[t] = __builtin_amdgcn_wmma_f32_16x16x4_f32(
                    false, a, false, b, (short)0, acc[t], false, false);
            }
        }

        // ---- spill C tiles (+bias) to LDS.  C layout: lane<16 -> M=r, N=lane;
        //      lane>=16 -> M=8+r, N=lane-16.
#pragma unroll
        for (int t = 0; t < NTILE; ++t) {
#pragma unroll
            for (int r = 0; r < 8; ++r) {
                KV[w][(ksel * 8 + r) * NCHP + 16 * t + mrow] = acc[t][r] + biasv[t];
            }
        }
        asm volatile("s_wait_dscnt 0" ::: "memory");   // intra-wave LDS phase fence

        // ---- scores + softmax: lane handles neighbor mrow for heads 2*ksel, 2*ksel+1
        const int h0 = 2 * ksel;
        float s0 = 0.f, s1 = 0.f;
#pragma unroll
        for (int d = 0; d < 8; ++d) {
            s0 += QB[w][h0 * 8 + d]     * KV[w][mrow * NCHP + h0 * 8 + d];
            s1 += QB[w][h0 * 8 + 8 + d] * KV[w][mrow * NCHP + h0 * 8 + 8 + d];
        }
        const float scale = 0.35355339059327373f;      // 1/sqrt(8)
        s0 *= scale; s1 *= scale;
        float m0 = s0, m1 = s1;
#pragma unroll
        for (int off = 8; off >= 1; off >>= 1) {       // stays within 16-lane half
            m0 = fmaxf(m0, __shfl_xor(m0, off, 32));
            m1 = fmaxf(m1, __shfl_xor(m1, off, 32));
        }
        float e0 = __expf(s0 - m0), e1 = __expf(s1 - m1);
        float t0 = e0, t1 = e1;
#pragma unroll
        for (int off = 8; off >= 1; off >>= 1) {
            t0 += __shfl_xor(t0, off, 32);
            t1 += __shfl_xor(t1, off, 32);
        }
        PB[w][mrow * NHEAD + h0]     = e0 / t0;
        PB[w][mrow * NHEAD + h0 + 1] = e1 / t1;
        asm volatile("s_wait_dscnt 0" ::: "memory");

        // ---- weighted sum of V: lane handles output channels lane and lane+32
        const int cA = lane, cB = lane + 32;
        const int hA = cA >> 4, hB = cB >> 4;
        float oA = 0.f, oB = 0.f;
#pragma unroll
        for (int m = 0; m < KNN; ++m) {
            oA += PB[w][m * NHEAD + hA] * KV[w][m * NCHP + 32 + cA];
            oB += PB[w][m * NHEAD + hB] * KV[w][m * NCHP + 32 + cB];
        }
        out[(size_t)n * DM + cA] = oA;
        out[(size_t)n * DM + cB] = oB;
        asm volatile("s_wait_dscnt 0" ::: "memory");   // keep next iter's stores behind reads
    }
}

extern "C" void kernel_launch(void* const* d_in, const int* in_sizes, int n_in,
                              void* d_out, int out_size, void* d_ws, size_t ws_size,
                              hipStream_t stream) {
    const float* coords0 = (const float*)d_in[0];
    const float* coords1 = (const float*)d_in[1];
    const float* feats0  = (const float*)d_in[2];
    const float* feats1  = (const float*)d_in[3];
    const int*   knn     = (const int*)  d_in[4];
    const float* Wq      = (const float*)d_in[5];
    const float* bq      = (const float*)d_in[6];
    const float* Wk      = (const float*)d_in[7];
    const float* bk      = (const float*)d_in[8];
    const float* Wv      = (const float*)d_in[9];
    const float* bv      = (const float*)d_in[10];
    float*       out     = (float*)d_out;

    const int n1 = in_sizes[1] / 3;

    // ~12500 waves total -> each wave grid-strides over ~8 points, amortizing
    // the per-block LDS weight staging over 32 points.
    int blocks = 3125;
    int max_blocks = (n1 + WPB - 1) / WPB;     // at least 1 point per wave
    if (blocks > max_blocks) blocks = max_blocks;
    if (blocks < 1) blocks = 1;

    knn_attn_kernel<<<blocks, BLOCK, 0, stream>>>(
        coords0, coords1, feats0, feats1, knn,
        Wq, bq, Wk, bk, Wv, bv, out, n1);

    // Second tuple element: pass-through knn_idxs (int32), appended after the
    // [n1, 64] float block.
    hipMemcpyAsync((char*)d_out + (size_t)n1 * DM * sizeof(float),
                   d_in[4], (size_t)in_sizes[4] * sizeof(int),
                   hipMemcpyDeviceToDevice, stream);
}